// Syn_Compact_Triplet_MMD_43284680409704
// MI455X (gfx1250) — compile-verified
//
#include <hip/hip_runtime.h>
#include <hip/hip_bf16.h>

// ---------------------------------------------------------------------------
// Problem constants (from reference)
// ---------------------------------------------------------------------------
#define NW_    4
#define STEP_  16           // 1 + 4 + 2 + 4 + 2 + 3
#define T_     512
#define D_     64
#define TD_    (T_ * D_)    // 32768
#define NPAIR_ 60           // NW * (STEP-1)
#define BIGF   1.0e9f
#define GAMMAF 5.0f

typedef __attribute__((ext_vector_type(2))) float v2f;
typedef __attribute__((ext_vector_type(8))) float v8f;

// ---------------------------------------------------------------------------
// Kernel 1: per-row squared norms for every (seq,row) of data
// rowsq[seq*T + t] = sum_d data[seq,t,d]^2       seq in [0,64)
// ---------------------------------------------------------------------------
__global__ __launch_bounds__(256)
void rowsq_kernel(const float* __restrict__ data, float* __restrict__ rowsq) {
    int row = blockIdx.x * 256 + threadIdx.x;          // 0 .. 32767
    const float4* p = (const float4*)(data + (size_t)row * D_);
    float s = 0.0f;
    #pragma unroll
    for (int k = 0; k < D_ / 4; ++k) {
        float4 v = p[k];
        s += v.x * v.x + v.y * v.y + v.z * v.z + v.w * v.w;
    }
    rowsq[row] = s;
}

// ---------------------------------------------------------------------------
// Kernel 2: cost[pair, t, s] = a2[t] + b2[s] - 2 * <anchor_t, other_s>
// One wave32 computes one 16x16 output tile with V_WMMA_F32_16X16X4_F32,
// looping K = 64 in steps of 4 (16 WMMA ops per tile).
// grid: 15360 blocks * 4 waves = 61440 tiles = 60 pairs * 32 * 32 tiles
// ---------------------------------------------------------------------------
__global__ __launch_bounds__(128)
void cost_gemm_kernel(const float* __restrict__ data,
                      const float* __restrict__ rowsq,
                      float* __restrict__ cost) {
    const int lane   = threadIdx.x & 31;
    const int wave   = threadIdx.x >> 5;
    const int tileId = blockIdx.x * 4 + wave;
    const int pair   = tileId >> 10;          // / (32*32)
    const int rem    = tileId & 1023;
    const int tm     = rem >> 5;
    const int tn     = rem & 31;
    const int w      = pair / 15;
    const int j      = pair % 15;

    const int lo = lane & 15;
    const int hi = lane >> 4;

    const float* A = data + (size_t)(w * STEP_) * TD_;            // anchors[w]
    const float* B = data + (size_t)(w * STEP_ + 1 + j) * TD_;    // others[w,j]
    const float* pa = A + (size_t)(tm * 16 + lo) * D_ + 2 * hi;
    const float* pb = B + (size_t)(tn * 16 + lo) * D_ + 2 * hi;

    v8f c = {};
    #pragma unroll
    for (int kk = 0; kk < D_; kk += 4) {
        v2f a = *(const v2f*)(pa + kk);
        v2f b = *(const v2f*)(pb + kk);
        c = __builtin_amdgcn_wmma_f32_16x16x4_f32(
                /*neg_a=*/false, a, /*neg_b=*/false, b,
                /*c_mod=*/(short)0, c, /*reuse_a=*/false, /*reuse_b=*/false);
    }

    const float* a2 = rowsq + (size_t)(w * STEP_) * T_;
    const float* b2 = rowsq + (size_t)(w * STEP_ + 1 + j) * T_;
    const float bn = b2[tn * 16 + lo];
    float* out = cost + (size_t)pair * T_ * T_;
    #pragma unroll
    for (int r = 0; r < 8; ++r) {
        int m = tm * 16 + r + 8 * hi;
        out[(size_t)m * T_ + tn * 16 + lo] = a2[m] + bn - 2.0f * c[r];
    }
}

// ---------------------------------------------------------------------------
// Kernel 3: soft-DTW anti-diagonal scan. One block (512 threads) per pair.
// Triple-buffered LDS rows; one barrier per diagonal.
// Software-pipelined: the global load of D[i, k+1-i] is issued before step
// k's LDS-dependent softmin, so its latency (L2-resident cost array, 192 MB
// global L2 > 63 MB working set) overlaps the serial exp/log chain instead
// of extending it.
// ---------------------------------------------------------------------------
__global__ __launch_bounds__(512)
void softdtw_kernel(const float* __restrict__ cost,
                    const int* __restrict__ lens,
                    float* __restrict__ sdt) {
    __shared__ float buf[3][T_];
    const int pair = blockIdx.x;
    const int w = pair / 15;
    const int j = pair % 15;
    const float* __restrict__ rowp = cost + (size_t)pair * T_ * T_
                                          + (size_t)threadIdx.x * T_;
    const int la = lens[w * STEP_];
    const int lb = lens[w * STEP_ + 1 + j];
    const int i = threadIdx.x;

    buf[0][i] = BIGF;
    buf[1][i] = BIGF;
    __syncthreads();

    int r2i = 0, r1i = 1, rci = 2;
    const float invG = 1.0f / GAMMAF;
    const int kcap = la + lb - 2;

    // prefetch diagonal k = 0 (clamped, branch-free)
    int jj = 0 - i;
    float dnext = rowp[(jj >= 0 && jj < T_) ? jj : 0];

    for (int k = 0; k < 2 * T_ - 1; ++k) {
        const float dcur = dnext;
        const bool  valid = (k - i >= 0) && (k - i < T_);

        // issue next diagonal's load before touching LDS (pipelined)
        int jn = k + 1 - i;
        dnext = rowp[(jn >= 0 && jn < T_) ? jn : 0];

        float left = buf[r1i][i];
        float up   = (i > 0) ? buf[r1i][i - 1] : BIGF;
        float diag = (i > 0) ? buf[r2i][i - 1] : BIGF;
        if (i == 0 && k == 0) diag = 0.0f;

        float mn = fminf(diag, fminf(up, left));
        float s  = __expf((mn - diag) * invG) +
                   __expf((mn - up)   * invG) +
                   __expf((mn - left) * invG);
        float softmin = mn - GAMMAF * __logf(s);
        float d = valid ? dcur : BIGF;
        float r = valid ? (d + softmin) : BIGF;
        buf[rci][i] = r;
        if (k == kcap && i == la - 1) sdt[pair] = r;
        __syncthreads();
        int t = r2i; r2i = r1i; r1i = rci; rci = t;
    }
}

// ---------------------------------------------------------------------------
// Kernel 4: MMD Gram matrices. For each of 9 worker pairs, G = total·totalᵀ
// where total = 22 rows of length 32768 (rows >= 22 zero-padded to 32).
// One wave per 16x16 tile (4 waves per pair), K looped in 128-float chunks:
// global_prefetch 4 KB ahead, then 32 WMMA ops per chunk.
// ---------------------------------------------------------------------------
__global__ __launch_bounds__(128)
void gram_kernel(const float* __restrict__ data, float* __restrict__ G) {
    const int PI[9] = {0, 0, 0, 1, 1, 2, 2, 3, 3};
    const int PJ[9] = {1, 2, 3, 2, 3, 1, 3, 1, 2};

    const int lane = threadIdx.x & 31;
    const int wv   = threadIdx.x >> 5;
    const int p    = blockIdx.x;            // 0..8
    const int ti   = wv >> 1;
    const int tj   = wv & 1;
    const int lo   = lane & 15;
    const int hi   = lane >> 4;

    const int ra = ti * 16 + lo;
    const int rb = tj * 16 + lo;
    const bool va = (ra < 22);
    const bool vb = (rb < 22);

    // groups[w][g] = data[(w*STEP + g) * T*D], g in [0, 11)
    const float* pa = data;
    const float* pb = data;
    if (va) {
        int seq = (ra < 11) ? (PI[p] * STEP_ + ra) : (PJ[p] * STEP_ + ra - 11);
        pa = data + (size_t)seq * TD_ + 2 * hi;
    }
    if (vb) {
        int seq = (rb < 11) ? (PI[p] * STEP_ + rb) : (PJ[p] * STEP_ + rb - 11);
        pb = data + (size_t)seq * TD_ + 2 * hi;
    }

    v8f c = {};
    const v2f z = {0.0f, 0.0f};
    for (int kc = 0; kc < TD_; kc += 128) {
        // stream 4 KB ahead of the consume point
        if (va && kc + 1024 < TD_) __builtin_prefetch(pa + kc + 1024, 0, 1);
        if (vb && kc + 1024 < TD_) __builtin_prefetch(pb + kc + 1024, 0, 1);
        #pragma unroll
        for (int kk = 0; kk < 128; kk += 4) {
            v2f a = va ? *(const v2f*)(pa + kc + kk) : z;
            v2f b = vb ? *(const v2f*)(pb + kc + kk) : z;
            c = __builtin_amdgcn_wmma_f32_16x16x4_f32(
                    false, a, false, b, (short)0, c, false, false);
        }
    }

    float* out = G + (size_t)p * 1024;       // 32x32 per pair
    #pragma unroll
    for (int r = 0; r < 8; ++r) {
        int m = ti * 16 + r + 8 * hi;
        out[m * 32 + tj * 16 + lo] = c[r];
    }
}

// ---------------------------------------------------------------------------
// Kernel 5: scalar epilogue (triplet loss + MMD) -> out[0]
// ---------------------------------------------------------------------------
__global__ void finalize_kernel(const float* __restrict__ G,
                                const float* __restrict__ sdt,
                                const int* __restrict__ lens,
                                float* __restrict__ out) {
    if (threadIdx.x != 0 || blockIdx.x != 0) return;

    // dist[w][j] = sdt / (la + lb)
    float dist[NW_][15];
    for (int w = 0; w < NW_; ++w) {
        float la = (float)lens[w * STEP_];
        for (int j = 0; j < 15; ++j) {
            float lb = (float)lens[w * STEP_ + 1 + j];
            dist[w][j] = sdt[w * 15 + j] / (la + lb);
        }
    }

    float total = 0.0f;
    for (int w = 0; w < NW_; ++w) {
        const float* dg = dist[w];          // 6 "group" distances
        const float* dn = dist[w] + 6;      // 9 "negative" distances
        float ca = 0.0f, cb = 0.0f;
        for (int a = 0; a < 6; ++a) ca += dg[a];
        ca /= 6.0f;
        for (int b = 0; b < 5; ++b) cb += dn[b];
        cb /= 5.0f;
        float lksum = 0.0f;
        int nz = 0;
        for (int a = 0; a < 6; ++a)
            for (int b = 0; b < 9; ++b) {
                float l = dg[a] + 1.0f - dn[b];   // MARGIN = 1.0
                l = l > 0.0f ? l : 0.0f;
                lksum += l;
                if (l != 0.0f) nz++;
            }
        float intra = 0.0f;
        for (int a = 0; a < 6; ++a) intra += dg[a] - ca;
        float inter = 1.0f - fabsf(ca - cb);      // BETA = 1.0
        inter = inter > 0.0f ? inter : 0.0f;
        float lv = lksum / (float)(nz + 1);
        total += lv + intra * 0.1f + inter * 0.1f; // P = R = 0.1
    }
    total *= 0.25f;                                // mean over NW

    // MMD per pair; mmds padded with zero then max.
    float mmax = 0.0f;
    for (int p = 0; p < 9; ++p) {
        const float* g = G + (size_t)p * 1024;
        float sq[22];
        for (int i = 0; i < 22; ++i) sq[i] = g[i * 32 + i];
        float l2sum = 0.0f;
        for (int i = 0; i < 22; ++i)
            for (int jx = 0; jx < 22; ++jx)
                l2sum += sq[i] + sq[jx] - 2.0f * g[i * 32 + jx];
        float bw = l2sum / (22.0f * 22.0f - 22.0f);
        bw *= 0.25f;                               // / KMUL^(KNUM/2) = /4
        float mmd = 0.0f;
        for (int i = 0; i < 11; ++i)
            for (int jx = 0; jx < 11; ++jx) {
                int ii[4] = {i, 11 + i, i, 11 + i};
                int jj[4] = {jx, 11 + jx, 11 + jx, jx};
                float sgn[4] = {1.0f, 1.0f, -1.0f, -1.0f};
                for (int q = 0; q < 4; ++q) {
                    float l2 = sq[ii[q]] + sq[jj[q]] - 2.0f * g[ii[q] * 32 + jj[q]];
                    float bwk = bw, s = 0.0f;
                    for (int k = 0; k < 5; ++k) { s += __expf(-l2 / bwk); bwk *= 2.0f; }
                    mmd += sgn[q] * s;
                }
            }
        mmd /= 121.0f;
        mmax = fmaxf(mmax, mmd);
    }
    out[0] = total + 0.01f * mmax;                 // ALPHA = 0.01
}

// ---------------------------------------------------------------------------
// Launch
// ---------------------------------------------------------------------------
extern "C" void kernel_launch(void* const* d_in, const int* in_sizes, int n_in,
                              void* d_out, int out_size, void* d_ws, size_t ws_size,
                              hipStream_t stream) {
    const float* data = (const float*)d_in[0];
    const int*   lens = (const int*)d_in[1];
    float*       out  = (float*)d_out;

    // workspace layout (floats)
    float* ws    = (float*)d_ws;
    float* cost  = ws;                               // 60*512*512 = 15,728,640
    float* rowsq = cost + (size_t)NPAIR_ * T_ * T_;  // 64*512 = 32,768
    float* G     = rowsq + (size_t)NW_ * STEP_ * T_; // 9*32*32 = 9,216
    float* sdt   = G + 9 * 1024;                     // 60

    rowsq_kernel<<<128, 256, 0, stream>>>(data, rowsq);
    cost_gemm_kernel<<<15360, 128, 0, stream>>>(data, rowsq, cost);
    softdtw_kernel<<<NPAIR_, 512, 0, stream>>>(cost, lens, sdt);
    gram_kernel<<<9, 128, 0, stream>>>(data, G);
    finalize_kernel<<<1, 32, 0, stream>>>(G, sdt, lens, out);
}